// Behavior_Specific_42863773614188
// MI455X (gfx1250) — compile-verified
//
#include <hip/hip_runtime.h>
#include <stdint.h>

typedef __attribute__((ext_vector_type(4))) float v4f;

#define B_  512
#define S_  512
#define H_  128
#define T_  4
#define N_  (B_ * S_)          // 262144 tokens
#define SCAN_BLK 512           // tokens per scan block
#define NBLK (N_ / SCAN_BLK)   // 512 scan blocks

// ---------------------------------------------------------------------------
// Zero the seq staging region out[t][0][:][:] for t=0..3 (1 MiB total).
// 65536 float4 -> 256 blocks x 256 threads.
__global__ void k_zero(float* __restrict__ out) {
    int idx = blockIdx.x * 256 + threadIdx.x;       // 0..65535
    int t   = idx >> 14;                            // 16384 float4 per type
    int r   = idx & 16383;
    v4f z = {0.f, 0.f, 0.f, 0.f};
    v4f* dst = (v4f*)(out + (size_t)t * ((size_t)B_ * S_ * H_));
    dst[r] = z;
}

// ---------------------------------------------------------------------------
// Per-block counts of each behavior type (512 tokens/block).
__global__ void k_count(const int* __restrict__ bt, unsigned* __restrict__ partial) {
    __shared__ unsigned cnt[T_];
    int tid = threadIdx.x;
    if (tid < T_) cnt[tid] = 0u;
    __syncthreads();
    int base = blockIdx.x * SCAN_BLK;
    #pragma unroll
    for (int k = 0; k < 2; ++k) {
        int v = bt[base + tid + k * 256];
        if (v >= 1 && v <= T_) atomicAdd(&cnt[v - 1], 1u);
    }
    __syncthreads();
    if (tid < T_) partial[tid * NBLK + blockIdx.x] = cnt[tid];
}

// ---------------------------------------------------------------------------
// Single block: exclusive scan of the 512 block partials per type + totals.
__global__ void k_scan(const unsigned* __restrict__ partial,
                       unsigned* __restrict__ blockOff,
                       unsigned* __restrict__ totals) {
    __shared__ unsigned sh[NBLK];
    int tid = threadIdx.x;                           // 512 threads
    for (int t = 0; t < T_; ++t) {
        unsigned v = partial[t * NBLK + tid];
        sh[tid] = v;
        __syncthreads();
        for (int off = 1; off < NBLK; off <<= 1) {
            unsigned add = (tid >= off) ? sh[tid - off] : 0u;
            __syncthreads();
            sh[tid] += add;
            __syncthreads();
        }
        blockOff[t * NBLK + tid] = sh[tid] - v;      // exclusive prefix
        if (tid == NBLK - 1) totals[t] = sh[NBLK - 1];
        __syncthreads();
    }
}

// ---------------------------------------------------------------------------
// Scatter: recompute local ranks via one packed-u64 scan (16 bits/type; the 4
// types are mutually exclusive per token and a block holds only 512 tokens),
// then copy the selected rows into out[t][0][slot][:].
__global__ void k_scatter(const float* __restrict__ embs,
                          const int* __restrict__ bt,
                          const unsigned* __restrict__ blockOff,
                          const unsigned* __restrict__ totals,
                          float* __restrict__ out) {
    __shared__ unsigned long long sh[SCAN_BLK];
    int tid = threadIdx.x;                           // 512 threads
    int i   = blockIdx.x * SCAN_BLK + tid;
    int v   = bt[i];
    bool valid = (v >= 1 && v <= T_);
    int t = valid ? (v - 1) : 0;
    unsigned long long contrib = valid ? (1ull << (16 * t)) : 0ull;
    sh[tid] = contrib;
    __syncthreads();
    for (int off = 1; off < SCAN_BLK; off <<= 1) {
        unsigned long long add = (tid >= off) ? sh[tid - off] : 0ull;
        __syncthreads();
        sh[tid] += add;
        __syncthreads();
    }
    if (valid) {
        unsigned long long excl = sh[tid] - contrib;
        unsigned local = (unsigned)((excl >> (16 * t)) & 0xFFFFull);
        unsigned r  = blockOff[t * NBLK + blockIdx.x] + local;   // global rank
        int nv   = (int)totals[t];
        int slot = S_ - nv + (int)r;                 // right-aligned slot
        if (slot >= 0) {                             // slot < S_ guaranteed (r < nv)
            const v4f* src = (const v4f*)(embs + (size_t)i * H_);
            v4f* dst = (v4f*)(out + ((size_t)t * B_ * S_ + (size_t)slot) * H_);
            #pragma unroll
            for (int k = 0; k < H_ / 4; ++k) {
                v4f val = __builtin_nontemporal_load(src + k);   // row read exactly once
                dst[k] = val;
            }
        }
    }
}

// ---------------------------------------------------------------------------
// Broadcast out[t][0] -> out[t][1..511]. Each block: one type, a 32-row seq
// tile (16 KiB) async-staged into LDS (ASYNCcnt-tracked b128s), then streamed
// to 32 batch slices with non-temporal B128 stores.
// Grid: 4 types x 16 s-tiles x 16 batch-groups = 1024 blocks x 256 threads.
__global__ void k_bcast(float* __restrict__ out) {
    const int t  = blockIdx.z;
    const int s0 = blockIdx.x * 32;                  // 32 seq rows per tile
    const int b0 = blockIdx.y * 32;                  // 32 batches per block
    __shared__ v4f sh[1024];                         // 32*128 floats = 16 KiB
    const int tid = threadIdx.x;

    const float* src = out + (size_t)t * ((size_t)B_ * S_ * H_) + (size_t)s0 * H_;

    // Async global->LDS staging (4 x b128 per lane), tracked by ASYNCcnt.
    #pragma unroll
    for (int k = 0; k < 4; ++k) {
        int e = tid + k * 256;                       // float4 index in tile
        unsigned lds_addr = (unsigned)(uintptr_t)(&sh[e]);           // low 32b = LDS offset
        unsigned long long gaddr = (unsigned long long)(uintptr_t)(src + (size_t)e * 4);
        asm volatile("global_load_async_to_lds_b128 %0, %1, off"
                     :: "v"(lds_addr), "v"(gaddr) : "memory");
    }
    asm volatile("s_wait_asynccnt 0x0" ::: "memory");
    __syncthreads();

    v4f r0 = sh[tid];
    v4f r1 = sh[tid + 256];
    v4f r2 = sh[tid + 512];
    v4f r3 = sh[tid + 768];

    // out[t][0] already holds the seq; skip bb==0 without a branch in the loop.
    const int bstart = (b0 == 0) ? 1 : b0;
    const int bend   = b0 + 32;
    for (int bb = bstart; bb < bend; ++bb) {
        v4f* dst = (v4f*)(out + ((size_t)t * B_ + (size_t)bb) * ((size_t)S_ * H_)
                              + (size_t)s0 * H_);
        __builtin_nontemporal_store(r0, dst + tid);
        __builtin_nontemporal_store(r1, dst + tid + 256);
        __builtin_nontemporal_store(r2, dst + tid + 512);
        __builtin_nontemporal_store(r3, dst + tid + 768);
    }
}

// ---------------------------------------------------------------------------
extern "C" void kernel_launch(void* const* d_in, const int* in_sizes, int n_in,
                              void* d_out, int out_size, void* d_ws, size_t ws_size,
                              hipStream_t stream) {
    const float* embs = (const float*)d_in[0];       // [B,S,H] fp32
    const int*   bt   = (const int*)d_in[1];         // [B,S] int32
    float* out = (float*)d_out;                      // [4,B,S,H] fp32

    unsigned* partial  = (unsigned*)d_ws;            // 4*512 u32
    unsigned* blockOff = partial + T_ * NBLK;        // 4*512 u32
    unsigned* totals   = blockOff + T_ * NBLK;       // 4 u32   (~16.4 KiB total)

    k_zero   <<<256, 256, 0, stream>>>(out);
    k_count  <<<NBLK, 256, 0, stream>>>(bt, partial);
    k_scan   <<<1, NBLK, 0, stream>>>(partial, blockOff, totals);
    k_scatter<<<NBLK, SCAN_BLK, 0, stream>>>(embs, bt, blockOff, totals, out);
    k_bcast  <<<dim3(16, 16, 4), 256, 0, stream>>>(out);
}